// COLoRALinear_13254269075667
// MI455X (gfx1250) — compile-verified
//
#include <hip/hip_runtime.h>

typedef float v2f __attribute__((ext_vector_type(2)));
typedef float v8f __attribute__((ext_vector_type(8)));
typedef int   v4i __attribute__((vector_size(16)));   // matches builtin's "int __vector(4)"

constexpr int B_ = 4, S_ = 2048, DIN = 4096, DOUT = 4096, R_ = 8, T_ = 8;
constexpr int M_ = B_ * S_;            // 8192 rows
constexpr float SCALING = 2.0f;        // LORA_ALPHA / R
constexpr int KLR = 80;                // low-rank K: 8 + 64 = 72, padded to 80

// ---------------- workspace layout (in floats) ----------------
constexpr size_t WS_XMEAN = 0;                       // [B, DIN]
constexpr size_t WS_LOGIT = 16384;                   // [B, T]
constexpr size_t WS_ROUT  = 16416;                   // [B, T]
constexpr size_t WS_CS    = 16448;                   // scalar
constexpr size_t WS_HID   = 16512;                   // [M, KLR]
constexpr size_t WS_W2    = WS_HID + (size_t)M_ * KLR; // [DOUT, KLR]

// ---------------- async copy helpers (CDNA5 async-to-LDS path) ----------------
__device__ __forceinline__ void async_copy_b128(const float* g, unsigned lds_byte_off) {
#if __has_builtin(__builtin_amdgcn_global_load_async_to_lds_b128)
    __builtin_amdgcn_global_load_async_to_lds_b128(
        (__attribute__((address_space(1))) v4i*)(uintptr_t)g,
        (__attribute__((address_space(3))) v4i*)(uintptr_t)lds_byte_off,
        0, 0);
#else
    asm volatile("global_load_async_to_lds_b128 %0, %1, off"
                 :: "v"(lds_byte_off), "v"((unsigned long long)(uintptr_t)g)
                 : "memory");
#endif
}

__device__ __forceinline__ void wait_async0() {
#if __has_builtin(__builtin_amdgcn_s_wait_asynccnt)
    __builtin_amdgcn_s_wait_asynccnt(0);
#else
    asm volatile("s_wait_asynccnt 0x0" ::: "memory");
#endif
}

// ---------------- 1: column mean over S ----------------
__global__ void colora_mean_kernel(const float* __restrict__ X, float* __restrict__ xmean) {
    int idx = blockIdx.x * blockDim.x + threadIdx.x;   // [0, B*DIN)
    int b = idx / DIN, d = idx % DIN;
    const float* p = X + (size_t)b * S_ * DIN + d;
    float s = 0.f;
    for (int ss = 0; ss < S_; ++ss) s += p[(size_t)ss * DIN];
    xmean[idx] = s * (1.0f / S_);
}

// ---------------- 2: routing logits ----------------
__global__ void colora_logits_kernel(const float* __restrict__ xmean,
                                     const float* __restrict__ temb,
                                     float* __restrict__ logits) {
    int b = blockIdx.x / T_, t = blockIdx.x % T_;
    __shared__ float red[256];
    float s = 0.f;
    for (int d = threadIdx.x; d < DIN; d += 256)
        s += xmean[b * DIN + d] * temb[t * DIN + d];
    red[threadIdx.x] = s; __syncthreads();
    for (int off = 128; off > 0; off >>= 1) {
        if ((int)threadIdx.x < off) red[threadIdx.x] += red[threadIdx.x + off];
        __syncthreads();
    }
    if (threadIdx.x == 0) logits[b * T_ + t] = red[0];
}

// ---------------- 3: softmax + collab scales ----------------
__global__ void colora_routing_kernel(const float* __restrict__ logits,
                                      const float* __restrict__ collab,
                                      float* __restrict__ routing_scaled,
                                      float* __restrict__ cs_out) {
    float cw = 1.0f / (1.0f + __expf(-collab[0]));
    if (threadIdx.x == 0) cs_out[0] = cw * SCALING;
    if ((int)threadIdx.x < B_) {
        int b = threadIdx.x;
        float mx = -1e30f;
        for (int t = 0; t < T_; ++t) mx = fmaxf(mx, logits[b * T_ + t]);
        float e[T_], sum = 0.f;
        for (int t = 0; t < T_; ++t) { e[t] = __expf(logits[b * T_ + t] - mx); sum += e[t]; }
        float sc = (1.0f - cw) * SCALING / sum;
        for (int t = 0; t < T_; ++t) routing_scaled[b * T_ + t] = e[t] * sc;
    }
}

// ---------------- 4: pack W2 = [shared_B | task_B | 0pad]  [DOUT, KLR] ----------------
__global__ void colora_packw2_kernel(const float* __restrict__ sB,
                                     const float* __restrict__ tB,
                                     float* __restrict__ W2) {
    int idx = blockIdx.x * 256 + threadIdx.x;
    if (idx >= DOUT * KLR) return;
    int o = idx / KLR, j = idx % KLR;
    float v = 0.f;
    if (j < R_) v = sB[o * R_ + j];
    else if (j < R_ + T_ * R_) {
        int t = (j - R_) / R_, r = (j - R_) % R_;
        v = tB[((size_t)t * DOUT + o) * R_ + r];
    }
    W2[idx] = v;
}

// ---------------- 5: hid = scaled low-rank projections  [M, KLR] ----------------
__global__ __launch_bounds__(128)
void colora_hid_kernel(const float* __restrict__ X, const float* __restrict__ sA,
                       const float* __restrict__ tA,
                       const float* __restrict__ routing_scaled,
                       const float* __restrict__ cs,
                       float* __restrict__ hid) {
    int row = blockIdx.x;        // [0, M)
    int j = threadIdx.x;         // [0, 128)
    if (j >= KLR) return;
    float v = 0.f;
    if (j < R_ + T_ * R_) {
        const float* arow = (j < R_) ? (sA + (size_t)j * DIN)
                                     : (tA + (size_t)(j - R_) * DIN);
        const float* xr = X + (size_t)row * DIN;
        float s = 0.f;
        for (int d = 0; d < DIN; d += 4) {
            s += xr[d] * arow[d] + xr[d + 1] * arow[d + 1]
               + xr[d + 2] * arow[d + 2] + xr[d + 3] * arow[d + 3];
        }
        int b = row / S_;
        float scale = (j < R_) ? cs[0] : routing_scaled[b * T_ + (j - R_) / R_];
        v = s * scale;
    }
    hid[(size_t)row * KLR + j] = v;
}

// ---------------- 6: main WMMA GEMM, LDS double-buffered via async copies ----------------
// out[m, n] = sum_k x[m,k]*W[n,k] + sum_j hid[m,j]*W2[n,j] + bias[n]
constexpr int KB    = 16;             // k-block (floats)
constexpr int LROW  = 20;             // padded LDS row stride (floats): 16B-aligned rows,
                                      // (20*r + c) % 64 covers all banks -> conflict-free b64 reads
constexpr int TILEF = 128 * LROW;     // 2560 floats per tile
// smem: [buf0 A | buf0 B | buf1 A | buf1 B], 4*2560*4 = 40 KB

__global__ __launch_bounds__(128)
void colora_gemm_kernel(const float* __restrict__ X, const float* __restrict__ W,
                        const float* __restrict__ bias,
                        const float* __restrict__ hid, const float* __restrict__ W2,
                        float* __restrict__ out) {
    __shared__ float smem[4 * TILEF];

    const int tid   = threadIdx.x;
    const int lane  = tid & 31;
    const int wave  = tid >> 5;
    const int laneq = lane & 15;     // M/N index within 16
    const int laneh = lane >> 4;     // K-half selector (0/1)
    const int wm = wave & 1, wn = wave >> 1;
    const int mblk = blockIdx.y * 128;
    const int nblk = blockIdx.x * 128;
    const int m0 = mblk + wm * 64;
    const int n0 = nblk + wn * 64;

    const float* Xblk = X + (size_t)mblk * DIN;
    const float* Wblk = W + (size_t)nblk * DIN;

    const unsigned lds0 = (unsigned)(uintptr_t)(&smem[0]);
    const int crow = tid >> 2;       // copy row base (0..31), +32*i
    const int ccol = tid & 3;        // copy chunk (16B) within row

    // issue one 128x16 tile: 4 async b128 copies per thread
    auto issue_tile = [&](const float* gbase, int kb, unsigned tile_byte_off) {
#pragma unroll
        for (int i = 0; i < 4; ++i) {
            const int row = crow + 32 * i;
            const float* g = gbase + (size_t)row * DIN + kb + ccol * 4;
            async_copy_b128(g, lds0 + tile_byte_off + (unsigned)(row * LROW + ccol * 4) * 4u);
        }
    };

    const v8f vzero = {0.f, 0.f, 0.f, 0.f, 0.f, 0.f, 0.f, 0.f};
    v8f acc[4][4];
#pragma unroll
    for (int mt = 0; mt < 4; ++mt)
#pragma unroll
        for (int nt = 0; nt < 4; ++nt) acc[mt][nt] = vzero;

    // prologue: fill buffer 0
    issue_tile(Xblk, 0, 0u);
    issue_tile(Wblk, 0, (unsigned)(TILEF * 4));
    wait_async0();
    __syncthreads();

    int cur = 0;
    for (int kb = 0; kb < DIN; kb += KB) {
        const int nxt = cur ^ 1;
        if (kb + KB < DIN) {
            issue_tile(Xblk, kb + KB, (unsigned)(nxt * 2 * TILEF * 4));
            issue_tile(Wblk, kb + KB, (unsigned)((nxt * 2 + 1) * TILEF * 4));
        }
        const int abase = cur * 2 * TILEF;
        const int bbase = abase + TILEF;
#pragma unroll
        for (int kk = 0; kk < KB; kk += 4) {
            v2f af[4], bf[4];
#pragma unroll
            for (int mt = 0; mt < 4; ++mt)
                af[mt] = *(const v2f*)&smem[abase + (wm * 64 + mt * 16 + laneq) * LROW + kk + 2 * laneh];
#pragma unroll
            for (int nt = 0; nt < 4; ++nt)
                bf[nt] = *(const v2f*)&smem[bbase + (wn * 64 + nt * 16 + laneq) * LROW + kk + 2 * laneh];
#pragma unroll
            for (int mt = 0; mt < 4; ++mt)
#pragma unroll
                for (int nt = 0; nt < 4; ++nt)
                    acc[mt][nt] = __builtin_amdgcn_wmma_f32_16x16x4_f32(
                        false, af[mt], false, bf[nt], (short)0, acc[mt][nt], false, false);
        }
        wait_async0();
        __syncthreads();
        cur = nxt;
    }

    // ---- low-rank tail: K = KLR (80) over hid/W2, direct global loads ----
    const float* hptr[4];
    const float* wptr[4];
#pragma unroll
    for (int mt = 0; mt < 4; ++mt)
        hptr[mt] = hid + (size_t)(m0 + mt * 16 + laneq) * KLR + 2 * laneh;
#pragma unroll
    for (int nt = 0; nt < 4; ++nt)
        wptr[nt] = W2 + (size_t)(n0 + nt * 16 + laneq) * KLR + 2 * laneh;

    for (int k = 0; k < KLR; k += 4) {
        v2f af[4], bf[4];
#pragma unroll
        for (int mt = 0; mt < 4; ++mt) af[mt] = *(const v2f*)(hptr[mt] + k);
#pragma unroll
        for (int nt = 0; nt < 4; ++nt) bf[nt] = *(const v2f*)(wptr[nt] + k);
#pragma unroll
        for (int mt = 0; mt < 4; ++mt)
#pragma unroll
            for (int nt = 0; nt < 4; ++nt)
                acc[mt][nt] = __builtin_amdgcn_wmma_f32_16x16x4_f32(
                    false, af[mt], false, bf[nt], (short)0, acc[mt][nt], false, false);
    }

    // ---- store: C layout VGPR i -> row (laneh*8 + i), col = laneq ----
#pragma unroll
    for (int mt = 0; mt < 4; ++mt) {
#pragma unroll
        for (int nt = 0; nt < 4; ++nt) {
            const int col  = n0 + nt * 16 + laneq;
            const float bv = bias[col];
            const int row0 = m0 + mt * 16 + laneh * 8;
#pragma unroll
            for (int i = 0; i < 8; ++i)
                out[(size_t)(row0 + i) * DOUT + col] = acc[mt][nt][i] + bv;
        }
    }
}

extern "C" void kernel_launch(void* const* d_in, const int* in_sizes, int n_in,
                              void* d_out, int out_size, void* d_ws, size_t ws_size,
                              hipStream_t stream) {
    const float* x        = (const float*)d_in[0];
    const float* W        = (const float*)d_in[1];
    const float* b        = (const float*)d_in[2];
    const float* shared_A = (const float*)d_in[3];
    const float* shared_B = (const float*)d_in[4];
    const float* task_A   = (const float*)d_in[5];
    const float* task_B   = (const float*)d_in[6];
    const float* task_emb = (const float*)d_in[7];
    const float* collab   = (const float*)d_in[8];
    float* out = (float*)d_out;

    float* ws      = (float*)d_ws;
    float* xmean   = ws + WS_XMEAN;
    float* logits  = ws + WS_LOGIT;
    float* routing = ws + WS_ROUT;
    float* cs      = ws + WS_CS;
    float* hid     = ws + WS_HID;
    float* W2      = ws + WS_W2;

    colora_mean_kernel<<<(B_ * DIN) / 256, 256, 0, stream>>>(x, xmean);
    colora_logits_kernel<<<B_ * T_, 256, 0, stream>>>(xmean, task_emb, logits);
    colora_routing_kernel<<<1, 32, 0, stream>>>(logits, collab, routing, cs);
    colora_packw2_kernel<<<(DOUT * KLR + 255) / 256, 256, 0, stream>>>(shared_B, task_B, W2);
    colora_hid_kernel<<<M_, 128, 0, stream>>>(x, shared_A, task_A, routing, cs, hid);

    dim3 grid(DOUT / 128, M_ / 128);   // (32, 64)
    colora_gemm_kernel<<<grid, 128, 0, stream>>>(x, W, b, hid, W2, out);
}